// SampledAttentionDecoder_24266565222726
// MI455X (gfx1250) — compile-verified
//
#include <hip/hip_runtime.h>
#include <hip/hip_bf16.h>
#include <math.h>

// ---------------------------------------------------------------------------
// SampledAttentionDecoder for MI455X (gfx1250, wave32).
// Roofline: dominant cost = streaming encoder_outputs (256 MB) once from HBM
// (~11us at 23.3 TB/s) using the algebraic identities
//   scores[b,s]  = (W1^T h_new[b]) . eo[b,s] + h_new[b].b1 + mask[b,s]
//   context[b]   = W2 (sum_s alpha[b,s] eo[b,s]) + b2
// with an online (flash) softmax + online argmax over score+gumbel.
//
// CDNA5 paths exercised:
//  * chained V_WMMA_F32_16X16X4_F32 for the per-tile score mat-vec (exact f32)
//  * GLOBAL_LOAD_ASYNC_TO_LDS_B128 double-buffered tile streaming (ASYNCcnt),
//    with s_wait_asynccnt pipelining (guarded by __has_builtin fallbacks)
//  * LDS row stride padded to 260 floats -> conflict-free ds_load_b64 A-frags
// ---------------------------------------------------------------------------

#define Hd      256
#define Bn      8
#define Sn      32768
#define NCHUNK  64            // chunks per batch
#define SCHUNK  (Sn / NCHUNK) // 512 rows per chunk
#define NTILES  (SCHUNK / 16) // 32 tiles of 16 rows
#define LROW    260           // padded LDS row stride (floats) -> no bank conflicts
#define NEGV    (-10000000000.0f)

// workspace layout (float units)
#define WS_HNEW 0            // 8*256
#define WS_Q    2048         // 8*256
#define WS_OFF  4096         // 8
#define WS_EV   4104         // 8 (int bits)
#define WS_PART 4112         // 8*64 slots of 260 floats: m,l,best_t,best_i,c[256]
#define SLOT    260

// output layout (float/int32 units)
#define OUT_RESULT 0         // (B,1,H) 2048
#define OUT_HNEW   2048      // (1,B,H) 2048
#define OUT_EV     4096      // (B,)    8
#define OUT_ATTN   4104      // (B,S)   262144
#define OUT_MASK   266248    // (B,1,S) 262144

typedef __attribute__((ext_vector_type(2))) float v2f;
typedef __attribute__((ext_vector_type(4))) float v4f;
typedef __attribute__((ext_vector_type(8))) float v8f;
typedef int v4i __attribute__((vector_size(16)));
typedef __attribute__((address_space(1))) v4i* gas_v4i_ptr;
typedef __attribute__((address_space(3))) v4i* las_v4i_ptr;

#if __has_builtin(__builtin_amdgcn_global_load_async_to_lds_b128)
#define HAVE_ASYNC 1
#else
#define HAVE_ASYNC 0
#endif

#if HAVE_ASYNC
#if __has_builtin(__builtin_amdgcn_s_wait_asynccnt)
#define WAIT_ASYNC(n) do { __builtin_amdgcn_s_wait_asynccnt(n); asm volatile("" ::: "memory"); } while (0)
#else
#define WAIT_ASYNC(n) asm volatile("s_wait_asynccnt %0" :: "i"(n) : "memory")
#endif
#else
#define WAIT_ASYNC(n) do { } while (0)
#endif

__device__ __forceinline__ float sigmoidf_(float x) { return 1.0f / (1.0f + __expf(-x)); }

// Stream one 16x256 tile (contiguous 16KB of eo) into padded LDS rows.
// 32 instructions, each moving 128 floats coalesced (32 lanes x b128).
__device__ __forceinline__ void fill_tile(const float* __restrict__ gbase,
                                          float* __restrict__ lbase, int lane)
{
#if HAVE_ASYNC
#pragma unroll
    for (int r = 0; r < 16; ++r) {
#pragma unroll
        for (int p = 0; p < 2; ++p) {
            const float* g = gbase + r * Hd + p * 128 + lane * 4;
            float*       l = lbase + r * LROW + p * 128 + lane * 4;
            __builtin_amdgcn_global_load_async_to_lds_b128(
                (gas_v4i_ptr)(void*)g, (las_v4i_ptr)(void*)l, 0, 0);
        }
    }
#else
#pragma unroll
    for (int r = 0; r < 16; ++r) {
#pragma unroll
        for (int p = 0; p < 2; ++p) {
            *(v4f*)(lbase + r * LROW + p * 128 + lane * 4) =
                *(const v4f*)(gbase + r * Hd + p * 128 + lane * 4);
        }
    }
#endif
}

// ---------------- Kernel A: GRU step + q = W1^T h_new + off = h_new.b1 ------
__global__ __launch_bounds__(256)
void gru_q_kernel(const float* __restrict__ last_hidden,
                  const float* __restrict__ dec_in,
                  const float* __restrict__ W1, const float* __restrict__ b1,
                  const float* __restrict__ Wih, const float* __restrict__ Whh,
                  const float* __restrict__ bih, const float* __restrict__ bhh,
                  float* __restrict__ ws)
{
    const int b = blockIdx.x;
    const int t = threadIdx.x;
    __shared__ float xs[Hd], hs[Hd], hn[Hd], red[Hd];

    xs[t] = dec_in[b * Hd + t];
    hs[t] = last_hidden[b * Hd + t];
    __syncthreads();

    float gir = bih[t], giz = bih[Hd + t], gin = bih[2 * Hd + t];
    float ghr = bhh[t], ghz = bhh[Hd + t], ghn = bhh[2 * Hd + t];
    for (int k = 0; k < Hd; ++k) {
        const float xv = xs[k], hv = hs[k];
        gir += Wih[(size_t)t * Hd + k] * xv;
        giz += Wih[(size_t)(Hd + t) * Hd + k] * xv;
        gin += Wih[(size_t)(2 * Hd + t) * Hd + k] * xv;
        ghr += Whh[(size_t)t * Hd + k] * hv;
        ghz += Whh[(size_t)(Hd + t) * Hd + k] * hv;
        ghn += Whh[(size_t)(2 * Hd + t) * Hd + k] * hv;
    }
    const float r = sigmoidf_(gir + ghr);
    const float z = sigmoidf_(giz + ghz);
    const float n = tanhf(gin + r * ghn);
    const float hnew = (1.0f - z) * n + z * hs[t];

    hn[t] = hnew;
    ws[WS_HNEW + b * Hd + t] = hnew;
    red[t] = hnew * b1[t];
    __syncthreads();

    // q[j] = sum_i W1[i][j] * hnew[i]   (W1^T h_new)
    float qv = 0.0f;
    for (int i = 0; i < Hd; ++i) qv += W1[(size_t)i * Hd + t] * hn[i];
    ws[WS_Q + b * Hd + t] = qv;

    // reduce off = h_new . b1
    for (int stride = 128; stride > 0; stride >>= 1) {
        if (t < stride) red[t] += red[t + stride];
        __syncthreads();
    }
    if (t == 0) ws[WS_OFF + b] = red[0];
}

// ---------------- Kernel B: WMMA scores + online softmax/argmax -------------
__global__ __launch_bounds__(32)
void attn_stream_kernel(const float* __restrict__ eo,
                        const float* __restrict__ mask,
                        const float* __restrict__ gum,
                        float* __restrict__ ws,
                        float* __restrict__ attn_out /* d_out + OUT_ATTN */)
{
    const int bid  = blockIdx.x;
    const int b    = bid / NCHUNK;
    const int c    = bid % NCHUNK;
    const int lane = threadIdx.x;            // 0..31, one wave per block
    const int m16  = lane & 15;              // A-matrix row owned by this lane
    const int h2   = (lane >> 4) << 1;       // K sub-offset: 0 (lanes 0-15) / 2 (16-31)

    __shared__ float ldsq[Hd];
    __shared__ float tb[2][16 * LROW];       // double-buffered padded tiles (~33 KB)

    const float* __restrict__ q   = ws + WS_Q + b * Hd;
    const float  off              = ws[WS_OFF + b];
    const float* __restrict__ eob = eo  + (size_t)b * Sn * Hd;
    const float* __restrict__ mkb = mask + (size_t)b * Sn;
    const float* __restrict__ gmb = gum  + (size_t)b * Sn;

    // stage q into LDS (same-address reads broadcast -> conflict-free)
#pragma unroll
    for (int j = 0; j < 8; ++j) ldsq[lane + 32 * j] = q[lane + 32 * j];

    float run_m = -INFINITY, run_l = 0.0f, best_t = -INFINITY;
    int   best_i = 0;
    float cacc[8];
#pragma unroll
    for (int j = 0; j < 8; ++j) cacc[j] = 0.0f;

    const int s_base = c * SCHUNK;

    // prologue: async-fill tile 0
    fill_tile(eob + (size_t)s_base * Hd, &tb[0][0], lane);

    for (int t = 0; t < NTILES; ++t) {
        const int cur = t & 1;
        const int s0  = s_base + t * 16;

        // issue async fill for the next tile into the other buffer
        if (t + 1 < NTILES)
            fill_tile(eob + (size_t)(s0 + 16) * Hd, &tb[cur ^ 1][0], lane);

        // ensure tile t is resident (async ops complete in order)
        if (t + 1 < NTILES) { WAIT_ASYNC(32); } else { WAIT_ASYNC(0); }

        const float* __restrict__ T = &tb[cur][0];

        // ---- scores for 16 rows via chained f32 WMMA over K=256 ----
        const float* __restrict__ arow = T + m16 * LROW + h2;
        const float* __restrict__ qh   = ldsq + h2;
        v8f d = {0.f, 0.f, 0.f, 0.f, 0.f, 0.f, 0.f, 0.f};
#pragma unroll 8
        for (int k = 0; k < Hd; k += 4) {
            v2f a  = *(const v2f*)(arow + k);   // ds_load_b64, conflict-free
            v2f bb = *(const v2f*)(qh + k);     // broadcast
            d = __builtin_amdgcn_wmma_f32_16x16x4_f32(
                    false, a, false, bb, (short)0, d, false, false);
        }

        // D layout: vgpr v, lanes 0-15 -> row v ; lanes 16-31 -> row v+8
        float sc[16];
#pragma unroll
        for (int v = 0; v < 8; ++v) {
            sc[v]     = __shfl(d[v], 0, 32);
            sc[v + 8] = __shfl(d[v], 16, 32);
        }

        // per-lane mask & gumbel for row = lane (lanes 0..15)
        float mval = 0.0f, gval = 0.0f;
        if (lane < 16) {
            mval = mkb[s0 + lane];
            const float u = gmb[s0 + lane];
            gval = -__logf(-__logf(u));
        }

        float tv[16], outv = 0.0f, tmax = -INFINITY;
#pragma unroll
        for (int r = 0; r < 16; ++r) {
            const float mr  = __shfl(mval, r, 32);
            const float gr  = __shfl(gval, r, 32);
            const float pre = sc[r] + off + mr;       // attn_outputs value
            outv = (lane == r) ? pre : outv;
            const float tt = pre + gr;                // + gumbel for softmax/argmax
            tv[r] = tt;
            tmax  = fmaxf(tmax, tt);
            if (tt > best_t) { best_t = tt; best_i = s0 + r; }
        }
        if (lane < 16)
            attn_out[(size_t)b * Sn + s0 + lane] = outv;

        // ---- online softmax update + weighted row accumulation ----
        const float new_m = fmaxf(run_m, tmax);
        const float scale = __expf(run_m - new_m);    // 0 when run_m == -inf
        run_l *= scale;
#pragma unroll
        for (int j = 0; j < 8; ++j) cacc[j] *= scale;
        run_m = new_m;

#pragma unroll
        for (int r = 0; r < 16; ++r) {
            const float w = __expf(tv[r] - new_m);
            run_l += w;
            const float* __restrict__ row = T + r * LROW + lane;
#pragma unroll
            for (int j = 0; j < 8; ++j) cacc[j] += w * row[32 * j]; // ds_load_b32
        }
    }

    // write partial slot
    float* slot = ws + WS_PART + (size_t)(b * NCHUNK + c) * SLOT;
    if (lane == 0) {
        slot[0] = run_m;
        slot[1] = run_l;
        slot[2] = best_t;
        ((int*)slot)[3] = best_i;
    }
#pragma unroll
    for (int j = 0; j < 8; ++j) slot[4 + lane + 32 * j] = cacc[j];
}

// ---------------- Kernel C: combine partials, W2/W3 projections -------------
__global__ __launch_bounds__(256)
void reduce_kernel(const float* __restrict__ W2, const float* __restrict__ b2,
                   const float* __restrict__ W3, const float* __restrict__ b3,
                   float* __restrict__ ws, float* __restrict__ out)
{
    const int b = blockIdx.x;
    const int t = threadIdx.x;
    const float* __restrict__ part = ws + WS_PART + (size_t)b * NCHUNK * SLOT;

    __shared__ float sM, sL;
    __shared__ int   sEV;
    __shared__ float ctx[Hd];
    __shared__ float cat[2 * Hd];

    if (t == 0) {
        float M = -INFINITY;
        for (int c = 0; c < NCHUNK; ++c) M = fmaxf(M, part[c * SLOT]);
        float L = 0.0f, bt = -INFINITY;
        int bi = 0;
        for (int c = 0; c < NCHUNK; ++c) {
            const float mc = part[c * SLOT];
            L += part[c * SLOT + 1] * __expf(mc - M);
            const float btc = part[c * SLOT + 2];
            const int   bic = ((const int*)part)[c * SLOT + 3];
            if (btc > bt) { bt = btc; bi = bic; }
        }
        sM = M; sL = L; sEV = bi;
    }
    __syncthreads();

    const float M = sM, invL = 1.0f / sL;
    float cs = 0.0f;
    for (int c = 0; c < NCHUNK; ++c)
        cs += part[c * SLOT + 4 + t] * __expf(part[c * SLOT] - M);
    ctx[t] = cs * invL;                         // sum_s alpha[s] * eo[s, t]
    __syncthreads();

    // context = W2 @ ctx + b2
    float cv = b2[t];
    for (int h = 0; h < Hd; ++h) cv += W2[(size_t)t * Hd + h] * ctx[h];
    cat[t]      = cv;
    cat[Hd + t] = ws[WS_HNEW + b * Hd + t];
    __syncthreads();

    // result = W3 @ [context ; h_new] + b3
    float rv = b3[t];
    for (int h = 0; h < 2 * Hd; ++h) rv += W3[(size_t)t * 2 * Hd + h] * cat[h];

    out[OUT_RESULT + b * Hd + t] = rv;
    out[OUT_HNEW   + b * Hd + t] = cat[Hd + t];
    if (t == 0) {
        ((int*)out)[OUT_EV + b] = sEV;
        ((int*)ws)[WS_EV + b]   = sEV;
    }
}

// ---------------- Kernel D: mask copy with NEG scatter ----------------------
__global__ __launch_bounds__(256)
void mask_kernel(const float* __restrict__ mask, const float* __restrict__ ws,
                 float* __restrict__ out)
{
    const int idx = blockIdx.x * 256 + threadIdx.x;   // 0 .. B*S-1
    const int b = idx >> 15;
    const int s = idx & (Sn - 1);
    const int ev = ((const int*)ws)[WS_EV + b];
    float v = mask[idx];
    if (s == ev) v = NEGV;
    out[OUT_MASK + idx] = v;
}

// ---------------------------------------------------------------------------
extern "C" void kernel_launch(void* const* d_in, const int* in_sizes, int n_in,
                              void* d_out, int out_size, void* d_ws, size_t ws_size,
                              hipStream_t stream)
{
    (void)in_sizes; (void)n_in; (void)out_size; (void)ws_size;
    const float* last_hidden = (const float*)d_in[0];
    const float* dec_in      = (const float*)d_in[1];
    const float* eo          = (const float*)d_in[2];
    const float* mask        = (const float*)d_in[3];
    const float* gum         = (const float*)d_in[4];
    const float* W1          = (const float*)d_in[5];
    const float* b1          = (const float*)d_in[6];
    const float* W2          = (const float*)d_in[7];
    const float* b2          = (const float*)d_in[8];
    const float* W3          = (const float*)d_in[9];
    const float* b3          = (const float*)d_in[10];
    const float* Wih         = (const float*)d_in[11];
    const float* Whh         = (const float*)d_in[12];
    const float* bih         = (const float*)d_in[13];
    const float* bhh         = (const float*)d_in[14];

    float* ws  = (float*)d_ws;
    float* out = (float*)d_out;

    gru_q_kernel<<<Bn, 256, 0, stream>>>(last_hidden, dec_in, W1, b1,
                                         Wih, Whh, bih, bhh, ws);
    attn_stream_kernel<<<Bn * NCHUNK, 32, 0, stream>>>(eo, mask, gum, ws,
                                                       out + OUT_ATTN);
    reduce_kernel<<<Bn, 256, 0, stream>>>(W2, b2, W3, b3, ws, out);
    mask_kernel<<<(Bn * Sn) / 256, 256, 0, stream>>>(mask, ws, out);
}